// MaskedLSTMNetworkWithMergedEmb_12541304504370
// MI455X (gfx1250) — compile-verified
//
#include <hip/hip_runtime.h>
#include <math.h>

// ---------------- CDNA5 WMMA types ----------------
typedef float v2f __attribute__((ext_vector_type(2)));
typedef float v8f __attribute__((ext_vector_type(8)));

#define M_TILE      64
#define LDS_STRIDE  260   // 256 + 4 pad: dword bank = 4*m + k pattern -> conflict-free b64 loads

static __device__ __forceinline__ v8f wmma4(v2f a, v2f b, v8f c) {
  // D = A(16x4 f32) * B(4x16 f32) + C(16x16 f32), full fp32 — matches reference numerics
  return __builtin_amdgcn_wmma_f32_16x16x4_f32(false, a, false, b, (short)0, c, false, false);
}

// ---------------- Kernel 1: feature assembly ----------------
// combined = [prof_p(8), prof_e(8), skill_p(12), skill_e(12),
//             effect_p(91), effect_e(91), cont(20)] -> 242, zero-padded to 256
__global__ void assemble_kernel(const float* __restrict__ inputs,
                                const float* __restrict__ prof_emb,
                                const float* __restrict__ skill_emb,
                                const float* __restrict__ eff_emb,
                                float* __restrict__ Xc, int Brows) {
  int r = blockIdx.x * blockDim.x + threadIdx.x;
  if (r >= Brows) return;
  const float* in = inputs + (size_t)r * 170;
  float* x = Xc + (size_t)r * 256;

  auto amax = [&](int s, int n) -> int {
    int best = 0; float bv = in[s];
    for (int i = 1; i < n; ++i) { float v = in[s + i]; if (v > bv) { bv = v; best = i; } }
    return best;
  };
  int pp = amax(4, 13);
  int pe = amax(80, 13);
  int sp = pp * 4 + amax(158, 4);
  int se = pe * 4 + amax(154, 4);

  int o = 0;
  for (int i = 0; i < 8; ++i)  x[o++] = prof_emb[pp * 8 + i];
  for (int i = 0; i < 8; ++i)  x[o++] = prof_emb[pe * 8 + i];
  for (int i = 0; i < 12; ++i) x[o++] = skill_emb[sp * 12 + i];
  for (int i = 0; i < 12; ++i) x[o++] = skill_emb[se * 12 + i];
  for (int j = 0; j < 13; ++j) {                    // effect_p, vals start 37
    for (int i = 0; i < 4; ++i) x[o++] = eff_emb[j * 4 + i];
    for (int i = 0; i < 3; ++i) x[o++] = in[37 + 3 * j + i];
  }
  for (int j = 0; j < 13; ++j) {                    // effect_e, vals start 113
    for (int i = 0; i < 4; ++i) x[o++] = eff_emb[j * 4 + i];
    for (int i = 0; i < 3; ++i) x[o++] = in[113 + 3 * j + i];
  }
  const int cidx[20] = {162,163,164,165, 17, 19,20,21, 22,
                        166,167,168,169, 93, 95,96,97, 98, 152,153};
  for (int i = 0; i < 20; ++i) x[o++] = in[cidx[i]];
  for (; o < 256; ++o) x[o] = 0.f;                  // pad K: 242..255
}

// ---------------- Kernel 2: weight repack ----------------
// All weights repacked K-pair-interleaved so a WMMA B-fragment is ONE b64 load:
//   wp[(p*N + n)*2 + j] = w[2p + j][n]   (p = K-pair index, j in {0,1})
// w0 is zero-padded from K=242 to K=256 here, killing the hot-loop guard.
// Gate weights are transposed on the fly: w_ihT[k][c] = w_ih[c][k].
__global__ void prep_kernel(const float* __restrict__ w0,
                            const float* __restrict__ w_ih, const float* __restrict__ w_hh,
                            const float* __restrict__ b_ih, const float* __restrict__ b_hh,
                            float* __restrict__ w0p,
                            float* __restrict__ wip, float* __restrict__ whp,
                            float* __restrict__ bsum) {
  int idx = blockIdx.x * blockDim.x + threadIdx.x;   // over 128*1024
  if (idx < 128 * 1024) {
    int p = idx >> 10, c = idx & 1023;               // pair p covers K rows 2p, 2p+1
    wip[(size_t)idx * 2 + 0] = w_ih[(size_t)c * 256 + 2 * p];
    wip[(size_t)idx * 2 + 1] = w_ih[(size_t)c * 256 + 2 * p + 1];
    whp[(size_t)idx * 2 + 0] = w_hh[(size_t)c * 256 + 2 * p];
    whp[(size_t)idx * 2 + 1] = w_hh[(size_t)c * 256 + 2 * p + 1];
  }
  if (idx < 128 * 256) {
    int p = idx >> 8, n = idx & 255;
    int k0 = 2 * p, k1 = 2 * p + 1;
    w0p[(size_t)idx * 2 + 0] = (k0 < 242) ? w0[(size_t)k0 * 256 + n] : 0.f;
    w0p[(size_t)idx * 2 + 1] = (k1 < 242) ? w0[(size_t)k1 * 256 + n] : 0.f;
  }
  if (idx < 1024) bsum[idx] = b_ih[idx] + b_hh[idx];
}

// w1 repack (same pair-interleave, K=256 exactly, done separately to keep grids simple)
__global__ void prep_w1_kernel(const float* __restrict__ w1, float* __restrict__ w1p) {
  int idx = blockIdx.x * blockDim.x + threadIdx.x;   // over 128*256
  if (idx >= 128 * 256) return;
  int p = idx >> 8, n = idx & 255;
  w1p[(size_t)idx * 2 + 0] = w1[(size_t)(2 * p)     * 256 + n];
  w1p[(size_t)idx * 2 + 1] = w1[(size_t)(2 * p + 1) * 256 + n];
}

// ---------------- Kernel 3: fused MLP(2) + LSTM gates + cell ----------------
// One workgroup = 64 batch rows; 8 waves; wave w owns M-subtile (w&3), N-half (w>>2).
__global__ void __launch_bounds__(256, 1)
fused_mlp_lstm_kernel(const float* __restrict__ Xc,
                      const float* __restrict__ h_in,
                      const float* __restrict__ c_in,
                      const float* __restrict__ w0p, const float* __restrict__ b0,
                      const float* __restrict__ w1p, const float* __restrict__ b1,
                      const float* __restrict__ wip, const float* __restrict__ whp,
                      const float* __restrict__ bsum,
                      float* __restrict__ h_out, float* __restrict__ c_out,
                      int Brows) {
  extern __shared__ float lds[];
  float* bufA = lds;                          // 64 x 260
  float* bufB = lds + M_TILE * LDS_STRIDE;    // 64 x 260

  const int tid   = threadIdx.x;
  const int lane  = tid & 31;
  const int wid   = tid >> 5;
  const int msub  = wid & 3;                  // 4 M-subtiles of 16 rows
  const int nhalf = wid >> 2;                 // 2 N-halves
  const int ln    = lane & 15;                // N / M index within fragment
  const int hh    = lane >> 4;                // lane half -> K pair select
  const size_t base = (size_t)blockIdx.x * M_TILE;
  const int arow = msub * 16 + ln;            // A-fragment row (M) for this lane

  const v2f* W0 = (const v2f*)w0p;            // [p*256 + n]
  const v2f* W1 = (const v2f*)w1p;
  const v2f* WI = (const v2f*)wip;            // [p*1024 + c]
  const v2f* WH = (const v2f*)whp;

  // stage Xc tile -> bufA (coalesced)
  for (int i = tid; i < M_TILE * 256; i += 256) {
    int rr = i >> 8, col = i & 255;
    size_t g = base + rr; if (g >= (size_t)Brows) g = Brows - 1;
    bufA[rr * LDS_STRIDE + col] = Xc[g * 256 + col];
  }
  __syncthreads();

  // ---- layer 0: h1 = relu(Xc @ w0 + b0), K padded to 256, branch-free ----
  for (int ni = 0; ni < 8; ++ni) {
    const int n0 = (nhalf * 8 + ni) * 16;
    v8f acc = {};
    #pragma unroll 4
    for (int kk = 0; kk < 64; ++kk) {
      const int p = kk * 2 + hh;              // K rows 2p, 2p+1
      v2f a = *(const v2f*)&bufA[arow * LDS_STRIDE + 2 * p];  // ds_load_b64
      v2f b = W0[p * 256 + n0 + ln];                          // global_load_b64
      acc = wmma4(a, b, acc);
    }
    const float bias = b0[n0 + ln];
    #pragma unroll
    for (int r = 0; r < 8; ++r) {
      float v = acc[r] + bias;
      bufB[(msub * 16 + r + 8 * hh) * LDS_STRIDE + n0 + ln] = v > 0.f ? v : 0.f;
    }
  }
  __syncthreads();

  // ---- layer 1: x2 = relu(h1 @ w1 + b1), K = 256 -> bufA ----
  for (int ni = 0; ni < 8; ++ni) {
    const int n0 = (nhalf * 8 + ni) * 16;
    v8f acc = {};
    #pragma unroll 4
    for (int kk = 0; kk < 64; ++kk) {
      const int p = kk * 2 + hh;
      v2f a = *(const v2f*)&bufB[arow * LDS_STRIDE + 2 * p];
      v2f b = W1[p * 256 + n0 + ln];
      acc = wmma4(a, b, acc);
    }
    const float bias = b1[n0 + ln];
    #pragma unroll
    for (int r = 0; r < 8; ++r) {
      float v = acc[r] + bias;
      bufA[(msub * 16 + r + 8 * hh) * LDS_STRIDE + n0 + ln] = v > 0.f ? v : 0.f;
    }
  }
  __syncthreads();

  // stage h_in tile -> bufB (h1 no longer needed)
  for (int i = tid; i < M_TILE * 256; i += 256) {
    int rr = i >> 8, col = i & 255;
    size_t g = base + rr; if (g >= (size_t)Brows) g = Brows - 1;
    bufB[rr * LDS_STRIDE + col] = h_in[g * 256 + col];
  }
  __syncthreads();

  // ---- gates = x2 @ w_ihT + h_in @ w_hhT + bsum ; then LSTM cell ----
  for (int ni = 0; ni < 8; ++ni) {
    const int n0 = (nhalf * 8 + ni) * 16;
    const int col = n0 + ln;
    v8f gi = {}, gf = {}, gg = {}, go = {};
    #pragma unroll 2
    for (int kk = 0; kk < 64; ++kk) {
      const int p = kk * 2 + hh;
      v2f ax = *(const v2f*)&bufA[arow * LDS_STRIDE + 2 * p];
      v2f ah = *(const v2f*)&bufB[arow * LDS_STRIDE + 2 * p];
      const v2f* wi = WI + p * 1024;
      const v2f* wh = WH + p * 1024;
      gi = wmma4(ax, wi[col      ], gi);
      gi = wmma4(ah, wh[col      ], gi);
      gf = wmma4(ax, wi[col + 256], gf);
      gf = wmma4(ah, wh[col + 256], gf);
      gg = wmma4(ax, wi[col + 512], gg);
      gg = wmma4(ah, wh[col + 512], gg);
      go = wmma4(ax, wi[col + 768], go);
      go = wmma4(ah, wh[col + 768], go);
    }
    const float bi = bsum[col], bf = bsum[col + 256],
                bg = bsum[col + 512], bo = bsum[col + 768];
    #pragma unroll
    for (int r = 0; r < 8; ++r) {
      size_t row = base + (size_t)(msub * 16 + r + 8 * hh);
      if (row >= (size_t)Brows) continue;
      float ci = c_in[row * 256 + col];
      float vi = 1.f / (1.f + expf(-(gi[r] + bi)));
      float vf = 1.f / (1.f + expf(-(gf[r] + bf)));
      float vg = tanhf(gg[r] + bg);
      float vo = 1.f / (1.f + expf(-(go[r] + bo)));
      float c  = vf * ci + vi * vg;
      float h  = vo * tanhf(c);
      c_out[row * 256 + col] = c;
      h_out[row * 256 + col] = h;
    }
  }
}

// ---------------- Kernel 4: logits + mask (wave32 per row, shuffle reduce) ----------------
__global__ void logits_kernel(const float* __restrict__ h_out,
                              const float* __restrict__ inputs,
                              const float* __restrict__ w_logits,
                              const float* __restrict__ b_logits,
                              float* __restrict__ out, int Brows) {
  int row  = (blockIdx.x * blockDim.x + threadIdx.x) >> 5;
  int lane = threadIdx.x & 31;
  if (row >= Brows) return;
  const float* h = h_out + (size_t)row * 256;
  float s0 = 0.f, s1 = 0.f, s2 = 0.f, s3 = 0.f;
  #pragma unroll
  for (int u = 0; u < 8; ++u) {
    int k = lane * 8 + u;
    float hv = h[k];
    s0 += hv * w_logits[k * 4 + 0];
    s1 += hv * w_logits[k * 4 + 1];
    s2 += hv * w_logits[k * 4 + 2];
    s3 += hv * w_logits[k * 4 + 3];
  }
  for (int off = 16; off > 0; off >>= 1) {
    s0 += __shfl_down(s0, off, 32);
    s1 += __shfl_down(s1, off, 32);
    s2 += __shfl_down(s2, off, 32);
    s3 += __shfl_down(s3, off, 32);
  }
  if (lane == 0) {
    const float* in = inputs + (size_t)row * 170;
    out[(size_t)row * 4 + 0] = s0 + b_logits[0] + (1.f - in[0]) * -1e10f;
    out[(size_t)row * 4 + 1] = s1 + b_logits[1] + (1.f - in[1]) * -1e10f;
    out[(size_t)row * 4 + 2] = s2 + b_logits[2] + (1.f - in[2]) * -1e10f;
    out[(size_t)row * 4 + 3] = s3 + b_logits[3] + (1.f - in[3]) * -1e10f;
  }
}

// ---------------- launch ----------------
extern "C" void kernel_launch(void* const* d_in, const int* in_sizes, int n_in,
                              void* d_out, int out_size, void* d_ws, size_t ws_size,
                              hipStream_t stream) {
  const float* inputs    = (const float*)d_in[0];
  const float* h_in      = (const float*)d_in[1];
  const float* c_in      = (const float*)d_in[2];
  const float* prof_emb  = (const float*)d_in[3];
  const float* skill_emb = (const float*)d_in[4];
  const float* eff_emb   = (const float*)d_in[5];
  const float* w0        = (const float*)d_in[6];
  const float* b0        = (const float*)d_in[7];
  const float* w1        = (const float*)d_in[8];
  const float* b1        = (const float*)d_in[9];
  const float* w_ih      = (const float*)d_in[10];
  const float* w_hh      = (const float*)d_in[11];
  const float* b_ih      = (const float*)d_in[12];
  const float* b_hh      = (const float*)d_in[13];
  const float* w_logits  = (const float*)d_in[14];
  const float* b_logits  = (const float*)d_in[15];

  const int Brows = in_sizes[0] / 170;

  float* out        = (float*)d_out;               // [logits(B*4) | h_out(B*256) | c_out(B*256)]
  float* out_logits = out;
  float* out_h      = out + (size_t)Brows * 4;
  float* out_c      = out_h + (size_t)Brows * 256;

  float* ws   = (float*)d_ws;
  float* Xc   = ws;                                // Brows*256
  float* w0p  = Xc  + (size_t)Brows * 256;         // 128*256*2 = 65536
  float* w1p  = w0p + 128 * 256 * 2;               // 65536
  float* wip  = w1p + 128 * 256 * 2;               // 128*1024*2 = 262144
  float* whp  = wip + 128 * 1024 * 2;              // 262144
  float* bsum = whp + 128 * 1024 * 2;              // 1024

  assemble_kernel<<<(Brows + 255) / 256, 256, 0, stream>>>(
      inputs, prof_emb, skill_emb, eff_emb, Xc, Brows);
  prep_kernel<<<(128 * 1024 + 255) / 256, 256, 0, stream>>>(
      w0, w_ih, w_hh, b_ih, b_hh, w0p, wip, whp, bsum);
  prep_w1_kernel<<<(128 * 256 + 255) / 256, 256, 0, stream>>>(w1, w1p);

  const size_t ldsBytes = (size_t)2 * M_TILE * LDS_STRIDE * sizeof(float); // 133,120 B < 320 KB
  fused_mlp_lstm_kernel<<<(Brows + M_TILE - 1) / M_TILE, 256, ldsBytes, stream>>>(
      Xc, h_in, c_in, w0p, b0, w1p, b1, wip, whp, bsum, out_h, out_c, Brows);

  logits_kernel<<<((size_t)Brows * 32 + 255) / 256, 256, 0, stream>>>(
      out_h, inputs, w_logits, b_logits, out_logits, Brows);
}